// ConvergeToTargetGNN_730144440899
// MI455X (gfx1250) — compile-verified
//
#include <hip/hip_runtime.h>

typedef float v2f __attribute__((ext_vector_type(2)));
typedef float v8f __attribute__((ext_vector_type(8)));

// ---------------- degree / normalization ----------------

__global__ void fill_ones_k(float* __restrict__ p, int n) {
    int i = blockIdx.x * blockDim.x + threadIdx.x;
    if (i < n) p[i] = 1.0f;   // self-loop contributes 1 to every node's degree
}

__global__ void deg_accum_k(const int* __restrict__ dst, float* __restrict__ deg, int E) {
    int e = blockIdx.x * blockDim.x + threadIdx.x;
    if (e < E) atomicAdd(&deg[dst[e]], 1.0f);
}

__global__ void rsqrt_k(float* __restrict__ p, int n) {
    int i = blockIdx.x * blockDim.x + threadIdx.x;
    if (i < n) p[i] = rsqrtf(p[i]);   // deg >= 1 always (self-loops)
}

// coef[e] = dinv[src[e]] * dinv[dst[e]]  (computed once, reused by 3 scatter passes)
__global__ void coef_k(const int* __restrict__ src, const int* __restrict__ dst,
                       const float* __restrict__ dinv, float* __restrict__ coef, int E) {
    int e = blockIdx.x * blockDim.x + threadIdx.x;
    if (e < E) coef[e] = dinv[src[e]] * dinv[dst[e]];
}

// ---------------- dense transform: H[N, NOUT] = X[N, KDIM] @ W[KDIM, NOUT] ----------------
// One wave32 per 16-node tile, V_WMMA_F32_16X16X4_F32 (exact f32 MMA).
// W is staged per-block into LDS, zero-padded to [KPAD x NW16], so B-fragment
// loads are unconditional ds_load_b32 (no exec-mask churn). A-fragment loads are
// unconditional global loads with a branchless index clamp; out-of-range K lanes
// hit zero-padded weight rows, so their (finite) values contribute nothing.

template<int KDIM, int NOUT>
__global__ void gemm_wmma_k(const float* __restrict__ X, const float* __restrict__ W,
                            float* __restrict__ H, int ntiles, int nrows) {
    constexpr int NT   = (NOUT + 15) / 16;
    constexpr int KT   = (KDIM + 3) / 4;
    constexpr int KPAD = KT * 4;
    constexpr int NW16 = NT * 16;

    __shared__ float sW[KPAD * NW16];

    // cooperative zero-padded weight stage (KPAD*NW16 is an exact multiple of 256)
    for (int idx = threadIdx.x; idx < KPAD * NW16; idx += blockDim.x) {
        const int k = idx / NW16;
        const int n = idx - k * NW16;
        sW[idx] = (k < KDIM && n < NOUT) ? W[k * NOUT + n] : 0.0f;
    }
    __syncthreads();

    const int lane = threadIdx.x & 31;
    const int wv   = blockIdx.x * (blockDim.x >> 5) + (threadIdx.x >> 5);
    if (wv >= ntiles) return;                  // wave-uniform: EXEC all-1s for WMMA
    const int row0 = wv * 16;
    const int m    = lane & 15;
    const int half = lane >> 4;

    const unsigned lim = (unsigned)nrows * KDIM - 1u;   // branchless A-tail clamp

    v8f acc[NT];
#pragma unroll
    for (int t = 0; t < NT; ++t) acc[t] = 0.0f;

#pragma unroll
    for (int kt = 0; kt < KT; ++kt) {
        const int ka = kt * 4 + half * 2;
        const unsigned i0 = (unsigned)(row0 + m) * KDIM + ka;
        v2f a;
        a.x = X[min(i0,      lim)];
        a.y = X[min(i0 + 1u, lim)];
#pragma unroll
        for (int nt = 0; nt < NT; ++nt) {
            const int col = nt * 16 + m;
            v2f b;
            b.x = sW[ka * NW16 + col];
            b.y = sW[(ka + 1) * NW16 + col];
            acc[nt] = __builtin_amdgcn_wmma_f32_16x16x4_f32(
                false, a, false, b, (short)0, acc[nt], false, false);
        }
    }

#pragma unroll
    for (int nt = 0; nt < NT; ++nt) {
#pragma unroll
        for (int i = 0; i < 8; ++i) {
            const int r   = row0 + i + half * 8;
            const int col = nt * 16 + m;
            if (col < NOUT) H[(size_t)r * NOUT + col] = acc[nt][i];
        }
    }
}

// ---------------- per-layer epilogue pieces ----------------

// agg[i][c] = h[i][c] * dinv[i]^2 + b[c]   (self-loop message + bias)
template<int C>
__global__ void init_agg_k(const float* __restrict__ h, const float* __restrict__ dinv,
                           const float* __restrict__ b, float* __restrict__ agg, int n) {
    int t = blockIdx.x * blockDim.x + threadIdx.x;
    if (t >= n * C) return;
    const int i  = t / C;
    const int ch = t - i * C;
    const float di = dinv[i];
    agg[t] = fmaf(h[t], di * di, b[ch]);
}

// 16 lanes per edge, 4 channels each: contiguous 256B gather per edge,
// 4 f32 atomics into the L2-resident accumulator
__global__ void scatter64_k(const int* __restrict__ src, const int* __restrict__ dst,
                            const float* __restrict__ coef, const float* __restrict__ h,
                            float* __restrict__ agg, int E) {
    const int t = blockIdx.x * blockDim.x + threadIdx.x;
    const int e = t >> 4;
    if (e >= E) return;
    const int cg = (t & 15) << 2;
    const int s = src[e];
    const int d = dst[e];
    const float c = coef[e];
    const float4 v = *reinterpret_cast<const float4*>(h + (size_t)s * 64 + cg);
    float* p = agg + (size_t)d * 64 + cg;
    atomicAdd(p + 0, v.x * c);
    atomicAdd(p + 1, v.y * c);
    atomicAdd(p + 2, v.z * c);
    atomicAdd(p + 3, v.w * c);
}

__global__ void scatter3_k(const int* __restrict__ src, const int* __restrict__ dst,
                           const float* __restrict__ coef, const float* __restrict__ h,
                           float* __restrict__ out, int E) {
    const int e = blockIdx.x * blockDim.x + threadIdx.x;
    if (e >= E) return;
    const int s = src[e];
    const int d = dst[e];
    const float c = coef[e];
    const float* hp = h + (size_t)s * 3;
    float* p = out + (size_t)d * 3;
    atomicAdd(p + 0, hp[0] * c);
    atomicAdd(p + 1, hp[1] * c);
    atomicAdd(p + 2, hp[2] * c);
}

__global__ void relu_k(float* __restrict__ p, int n) {
    int i = blockIdx.x * blockDim.x + threadIdx.x;
    if (i < n) p[i] = fmaxf(p[i], 0.0f);
}

// ---------------- launcher ----------------

static inline int cdiv(long long a, int b) { return (int)((a + b - 1) / b); }

extern "C" void kernel_launch(void* const* d_in, const int* in_sizes, int n_in,
                              void* d_out, int out_size, void* d_ws, size_t ws_size,
                              hipStream_t stream) {
    const float* x   = (const float*)d_in[0];
    const int*   ei  = (const int*)d_in[1];
    const float* W1  = (const float*)d_in[2];
    const float* b1  = (const float*)d_in[3];
    const float* W2  = (const float*)d_in[4];
    const float* b2  = (const float*)d_in[5];
    const float* W3  = (const float*)d_in[6];
    const float* b3  = (const float*)d_in[7];
    float* out = (float*)d_out;

    const int N = in_sizes[0] / 9;     // 100000
    const int E = in_sizes[1] / 2;     // 1600000
    const int* src = ei;
    const int* dst = ei + E;

    float* dinv = (float*)d_ws;                       // N floats
    float* coef = dinv + N;                           // E floats
    float* h    = coef + E;                           // N*64 floats
    float* agg  = h + (size_t)N * 64;                 // N*64 floats
    (void)ws_size; (void)n_in; (void)out_size;

    const int ntiles = (N + 15) / 16;                 // 6250

    // --- symmetric normalization coefficients ---
    fill_ones_k<<<cdiv(N, 256), 256, 0, stream>>>(dinv, N);
    deg_accum_k<<<cdiv(E, 256), 256, 0, stream>>>(dst, dinv, E);
    rsqrt_k<<<cdiv(N, 256), 256, 0, stream>>>(dinv, N);
    coef_k<<<cdiv(E, 256), 256, 0, stream>>>(src, dst, dinv, coef, E);

    // --- layer 1: 9 -> 64, ReLU ---
    gemm_wmma_k<9, 64><<<cdiv(ntiles, 8), 256, 0, stream>>>(x, W1, h, ntiles, N);
    init_agg_k<64><<<cdiv((long long)N * 64, 256), 256, 0, stream>>>(h, dinv, b1, agg, N);
    scatter64_k<<<cdiv((long long)E * 16, 256), 256, 0, stream>>>(src, dst, coef, h, agg, E);
    relu_k<<<cdiv((long long)N * 64, 256), 256, 0, stream>>>(agg, N * 64);

    // --- layer 2: 64 -> 64, ReLU ---
    gemm_wmma_k<64, 64><<<cdiv(ntiles, 8), 256, 0, stream>>>(agg, W2, h, ntiles, N);
    init_agg_k<64><<<cdiv((long long)N * 64, 256), 256, 0, stream>>>(h, dinv, b2, agg, N);
    scatter64_k<<<cdiv((long long)E * 16, 256), 256, 0, stream>>>(src, dst, coef, h, agg, E);
    relu_k<<<cdiv((long long)N * 64, 256), 256, 0, stream>>>(agg, N * 64);

    // --- layer 3: 64 -> 3, no activation, writes d_out ---
    gemm_wmma_k<64, 3><<<cdiv(ntiles, 8), 256, 0, stream>>>(agg, W3, h, ntiles, N);
    init_agg_k<3><<<cdiv((long long)N * 3, 256), 256, 0, stream>>>(h, dinv, b3, out, N);
    scatter3_k<<<cdiv(E, 256), 256, 0, stream>>>(src, dst, coef, h, out, E);
}